// LocalLaplacianMessagePassing_59906203844896
// MI455X (gfx1250) — compile-verified
//
#include <hip/hip_runtime.h>
#include <hip/hip_bf16.h>

// LocalLaplacianMessagePassing for MI455X (gfx1250, wave32).
//
// Math: out rows of block b all equal  val[b] = S * sum_j w[b,j] * x[b,j,:]
// where w = P @ 1 and P = Phi Phi^T is the spectral projector onto the 16
// lowest eigenvectors of the normalized Laplacian L (32x32 per block).
// P is computed WITHOUT an eigensolver:
//   mu  : strict separator of lambda_15 / lambda_16 via inertia bisection
//         (negative-pivot count of LDL^T of L - mu*I)
//   P   = (I - sign(L - mu*I)) / 2, sign via Newton-Schulz X <- 1.5X - 0.5X^3
// Newton-Schulz is pure 32x32 fp32 GEMM -> V_WMMA_F32_16X16X4_F32 tiles.
//
// ws layout (floats): [0, Bc*1024) adjacency A ; [Bc*1024, +Bc*32) w vectors.

#define S_NODES 32
#define F_FEAT  256
#define LDSD    33          // LDS row stride (33 floats) to dodge bank conflicts
#define NS_ITERS 36
#define BIS_ITERS 24

typedef __attribute__((ext_vector_type(2))) float v2f;
typedef __attribute__((ext_vector_type(8))) float v8f;

// C += A(16x16) * B(16x16), both in LDS row-major with stride LDSD.
// V_WMMA_F32_16X16X4_F32 layouts (wave32):
//   A (16x4): element (m,k): lane = m + 16*(k>>1), reg = k&1
//   B (4x16): element (k,n): lane = n + 16*(k>>1), reg = k&1
//   C (16x16): element (M,n): lane = n + 16*(M>=8 in upper half), reg g: M = g + 8*(lane>=16)
__device__ __forceinline__ v8f llmp_mm16(const float* Abase, const float* Bbase,
                                         v8f acc, int ln16, int kh) {
#pragma unroll
  for (int kc = 0; kc < 16; kc += 4) {
    v2f av, bv;
    av.x = Abase[ln16 * LDSD + kc + kh];
    av.y = Abase[ln16 * LDSD + kc + kh + 1];
    bv.x = Bbase[(kc + kh) * LDSD + ln16];
    bv.y = Bbase[(kc + kh + 1) * LDSD + ln16];
    acc = __builtin_amdgcn_wmma_f32_16x16x4_f32(false, av, false, bv,
                                                (short)0, acc, false, false);
  }
  return acc;
}

__global__ void llmp_zero(float* __restrict__ p, int n) {
  int i = blockIdx.x * 256 + threadIdx.x;
  if (i < n) p[i] = 0.0f;
}

__global__ void llmp_scatter(const int* __restrict__ ei, int etot,
                             float* __restrict__ A) {
  int e = blockIdx.x * 256 + threadIdx.x;
  if (e < etot) {
    int sN = ei[e];
    int dN = ei[etot + e];
    atomicAdd(&A[((size_t)(sN >> 5)) * 1024 + ((sN & 31) << 5) + (dN & 31)], 1.0f);
  }
}

__global__ __launch_bounds__(32) void llmp_solve(const float* __restrict__ Ag,
                                                 float* __restrict__ wOut) {
  __shared__ float sL[S_NODES * LDSD];   // normalized Laplacian
  __shared__ float sW[S_NODES * LDSD];   // LDL scratch, later X^2
  __shared__ float sX[S_NODES * LDSD];   // Newton-Schulz iterate
  __shared__ float sdinv[S_NODES];

  const int b    = blockIdx.x;
  const int lane = threadIdx.x;          // 0..31, one wave
  const float* A = Ag + (size_t)b * 1024;

  // ---- load A (row per lane), degree, dinv ----
  float deg = 0.0f;
  for (int j = 0; j < 32; ++j) {
    float a = A[lane * 32 + j];
    sL[lane * LDSD + j] = a;
    deg += a;
  }
  sdinv[lane] = (deg > 0.0f) ? rsqrtf(deg) : 0.0f;
  __syncthreads();

  // ---- L = I - dinv A dinv ----
  {
    float di = sdinv[lane];
    for (int j = 0; j < 32; ++j) {
      float v = -di * sL[lane * LDSD + j] * sdinv[j];
      if (j == lane) v += 1.0f;
      sL[lane * LDSD + j] = v;
    }
  }
  __syncthreads();

  // ---- inertia bisection: mu with #(lambda < mu) == 16 ----
  float lo = -1.0e-3f, hi = 2.001f;
  for (int itb = 0; itb < BIS_ITERS; ++itb) {
    float mu = 0.5f * (lo + hi);
    for (int j = 0; j < 32; ++j) {
      float v = sL[lane * LDSD + j];
      if (j == lane) v -= mu;
      sW[lane * LDSD + j] = v;
    }
    __syncthreads();
    int negs = 0;
    for (int k = 0; k < 32; ++k) {
      float piv = sW[k * LDSD + k];               // uniform across lanes
      if (piv < 0.0f) negs++;
      float apiv = (fabsf(piv) < 1e-20f) ? (piv < 0.0f ? -1e-20f : 1e-20f) : piv;
      int i = k + 1 + lane;                        // one trailing row per lane
      if (i < 32) {
        float f = sW[i * LDSD + k] / apiv;
        for (int j = k; j < 32; ++j) sW[i * LDSD + j] -= f * sW[k * LDSD + j];
      }
      __syncthreads();
    }
    if (negs >= 16) hi = mu; else lo = mu;
  }
  const float mu = 0.5f * (lo + hi);

  // ---- X0 = (L - mu I) / s, spectrum in (-1, 1) ----
  const float inv_s = 1.0f / 2.2f;
  for (int j = 0; j < 32; ++j) {
    float v = sL[lane * LDSD + j];
    if (j == lane) v -= mu;
    sX[lane * LDSD + j] = v * inv_s;
  }
  __syncthreads();

  // ---- Newton-Schulz: X <- 1.5 X - 0.5 X^3 (WMMA f32 16x16x4) ----
  const int ln16  = lane & 15;
  const int kh    = (lane >> 4) << 1;   // 0 or 2 (K half-select)
  const int cbase = (lane >> 4) << 3;   // 0 or 8 (C row base)

  for (int it = 0; it < NS_ITERS; ++it) {
    // sW = X @ X
    for (int r = 0; r < 2; ++r)
      for (int c = 0; c < 2; ++c) {
        v8f acc = {};
        acc = llmp_mm16(&sX[(r * 16) * LDSD + 0],  &sX[0 * LDSD + c * 16],  acc, ln16, kh);
        acc = llmp_mm16(&sX[(r * 16) * LDSD + 16], &sX[16 * LDSD + c * 16], acc, ln16, kh);
#pragma unroll
        for (int g = 0; g < 8; ++g)
          sW[(r * 16 + cbase + g) * LDSD + c * 16 + ln16] = acc[g];
      }
    __syncthreads();

    // X = 1.5 X - 0.5 (sW @ X), one column block at a time
    for (int c = 0; c < 2; ++c) {
      v8f t0 = {}, t1 = {};
      t0 = llmp_mm16(&sW[0],              &sX[0 * LDSD + c * 16],  t0, ln16, kh);
      t0 = llmp_mm16(&sW[16],             &sX[16 * LDSD + c * 16], t0, ln16, kh);
      t1 = llmp_mm16(&sW[16 * LDSD + 0],  &sX[0 * LDSD + c * 16],  t1, ln16, kh);
      t1 = llmp_mm16(&sW[16 * LDSD + 16], &sX[16 * LDSD + c * 16], t1, ln16, kh);
      __syncthreads();
#pragma unroll
      for (int g = 0; g < 8; ++g) {
        int r0 = cbase + g, col = c * 16 + ln16;
        float x0 = sX[r0 * LDSD + col];
        sX[r0 * LDSD + col] = 1.5f * x0 - 0.5f * t0[g];
        float x1 = sX[(16 + r0) * LDSD + col];
        sX[(16 + r0) * LDSD + col] = 1.5f * x1 - 0.5f * t1[g];
      }
      __syncthreads();
    }
  }

  // ---- w = 0.5 * (1 - X @ ones) ----
  float v = 0.0f;
  for (int j = 0; j < 32; ++j) v += sX[lane * LDSD + j];
  wOut[(size_t)b * 32 + lane] = 0.5f * (1.0f - v);
}

__global__ __launch_bounds__(256) void llmp_out(const float* __restrict__ x,
                                                const float* __restrict__ w,
                                                float* __restrict__ out) {
  const int b = blockIdx.x;
  const int f = threadIdx.x;             // 256 features
  __shared__ float sw[32];
  if (threadIdx.x < 32) sw[threadIdx.x] = w[(size_t)b * 32 + threadIdx.x];
  __syncthreads();
  const float* xb = x + (size_t)b * S_NODES * F_FEAT;
  float acc = 0.0f;
#pragma unroll
  for (int j = 0; j < 32; ++j) acc += sw[j] * xb[j * F_FEAT + f];
  acc *= (float)S_NODES;
  float* ob = out + (size_t)b * S_NODES * F_FEAT;
#pragma unroll
  for (int j = 0; j < 32; ++j) ob[j * F_FEAT + f] = acc;
}

extern "C" void kernel_launch(void* const* d_in, const int* in_sizes, int n_in,
                              void* d_out, int out_size, void* d_ws, size_t ws_size,
                              hipStream_t stream) {
  const float* x  = (const float*)d_in[0];
  const int*   ei = (const int*)d_in[1];
  const int Nn   = in_sizes[0] / F_FEAT;   // 131072
  const int Bc   = Nn / S_NODES;           // 4096
  const int etot = in_sizes[1] / 2;        // 524288

  float* wsA = (float*)d_ws;               // Bc*1024 floats (16 MB)
  float* wsW = wsA + (size_t)Bc * 1024;    // Bc*32 floats

  const int nA = Bc * 1024;
  llmp_zero<<<(nA + 255) / 256, 256, 0, stream>>>(wsA, nA);
  llmp_scatter<<<(etot + 255) / 256, 256, 0, stream>>>(ei, etot, wsA);
  llmp_solve<<<Bc, 32, 0, stream>>>(wsA, wsW);
  llmp_out<<<Bc, 256, 0, stream>>>(x, wsW, (float*)d_out);
}